// KalmanFilter_26079041421461
// MI455X (gfx1250) — compile-verified
//
#include <hip/hip_runtime.h>
#include <hip/hip_bf16.h>

typedef __attribute__((ext_vector_type(2))) float v2f;
typedef __attribute__((ext_vector_type(8))) float v8f;

namespace {
constexpr int NS  = 32;    // state dim
constexpr int MD  = 8;     // control dim
constexpr int PD  = 16;    // obs dim
constexpr int TT  = 128;   // time steps
constexpr int NB  = 64;    // batch
constexpr int NTH = 128;   // threads/block = 4 wave32
constexpr int NWV = 4;
constexpr float QV = 0.01f;  // STD_DYN^2
constexpr float RV = 0.01f;  // STD_OBS^2
}

// Prefetch [p, p+bytes) into cache, one 128B line per thread slot.
// Lowers to global_prefetch_b8 on gfx1250 (speculative, no counters).
__device__ __forceinline__ void prefetch_bytes(const void* p, int bytes, int tid)
{
  const char* c = (const char*)p;
  for (int off = tid * 128; off < bytes; off += NTH * 128)
    __builtin_prefetch(c + off, 0, 1);
}

// D(Mrows x Ncols, ldd) = X(Mrows x Kdim, ldx) * Yop
// Yop = transY ? Y^T (Y stored Ncols x Kdim, ldy) : Y (Kdim x Ncols, ldy)
// Uses V_WMMA_F32_16X16X4_F32; tiles distributed across the 4 waves.
__device__ __forceinline__ void wmma_mm(const float* X, int ldx,
                                        const float* Y, int ldy, bool transY,
                                        float* D, int ldd,
                                        int Mrows, int Ncols, int Kdim, int tid)
{
  const int wave = tid >> 5;
  const int lane = tid & 31;
  const int half = lane >> 4;     // lanes 16-31 hold the upper K pair / rows M+8
  const int l16  = lane & 15;
  const int nti = Mrows >> 4, ntj = Ncols >> 4;
  const int ntiles = nti * ntj;
  for (int tile = wave; tile < ntiles; tile += NWV) {
    const int ti = tile / ntj, tj = tile % ntj;
    const int rowA = ti * 16 + l16;
    const int colB = tj * 16 + l16;
    v8f c = {};
    for (int k = 0; k < Kdim; k += 4) {
      const int kk = k + 2 * half;
      v2f a, b;
      // f32 A-fragment 16x4: lanes0-15 -> K=k,k+1 ; lanes16-31 -> K=k+2,k+3
      a.x = X[rowA * ldx + kk];
      a.y = X[rowA * ldx + kk + 1];
      if (!transY) {
        b.x = Y[kk * ldy + colB];
        b.y = Y[(kk + 1) * ldy + colB];
      } else {
        b.x = Y[colB * ldy + kk];
        b.y = Y[colB * ldy + kk + 1];
      }
      c = __builtin_amdgcn_wmma_f32_16x16x4_f32(false, a, false, b,
                                                (short)0, c, false, false);
    }
    const int baseRow = ti * 16 + 8 * half;   // C/D layout: VGPR r -> row r (+8 hi lanes)
#pragma unroll
    for (int r = 0; r < 8; ++r)
      D[(baseRow + r) * ldd + tj * 16 + l16] = c[r];
  }
}

// In-place Cholesky (lower triangle valid) of n x n SPD matrix in LDS.
// All NTH threads must call (contains barriers).
__device__ __forceinline__ void chol_inplace(float* M, int n, int tid)
{
  for (int k = 0; k < n; ++k) {
    if (tid == 0) M[k * n + k] = sqrtf(M[k * n + k]);
    __syncthreads();
    const float dk = M[k * n + k];
    for (int i = k + 1 + tid; i < n; i += NTH) M[i * n + k] /= dk;
    __syncthreads();
    const int rem = n - k - 1;
    for (int idx = tid; idx < rem * rem; idx += NTH) {
      const int i = k + 1 + idx / rem;
      const int j = k + 1 + idx % rem;
      if (j <= i) M[i * n + j] -= M[i * n + k] * M[j * n + k];
    }
    __syncthreads();
  }
}

extern "C" __global__ void __launch_bounds__(NTH)
kalman_smooth_kernel(const float* __restrict__ Yg, const float* __restrict__ Ug,
                     const float* __restrict__ Ag, const float* __restrict__ Bg,
                     const float* __restrict__ Cg, const float* __restrict__ mu0,
                     const float* __restrict__ Sig0, float* __restrict__ out)
{
  const int b   = blockIdx.x;
  const int tid = threadIdx.x;

  __shared__ float s_Sig[NS * NS];    // running Sigma (filtered) / Sf in backward
  __shared__ float s_A[NS * NS];
  __shared__ float s_Bm[NS * MD];
  __shared__ float s_C[PD * NS];
  __shared__ float s_T1[NS * NS];     // scratch (A@Sig, chol L, final JDJ^T...)
  __shared__ float s_Sigp[NS * NS];   // predicted Sigma
  __shared__ float s_IKC[NS * NS];    // I - K C  (fwd) / solve result W (bwd)
  __shared__ float s_T3[NS * NS];     // IKC@Sigp (fwd) / G then J (bwd)
  __shared__ float s_KKT[NS * NS];    // K K^T (fwd) / smoothed Sigma (bwd)
  __shared__ float s_JD[NS * NS];     // J @ D (bwd)
  __shared__ float s_T2[PD * NS];     // C @ Sigp
  __shared__ float s_X16[PD * NS];    // 16x32 solve result
  __shared__ float s_K[NS * PD];      // Kalman gain
  __shared__ float s_S[PD * PD];      // innovation covariance
  __shared__ float s_mu[NS];
  __shared__ float s_mup[NS];
  __shared__ float s_mus[NS];
  __shared__ float s_mp1[NS];
  __shared__ float s_y[PD];
  __shared__ float s_u[MD];
  __shared__ float s_r[PD];

  // ---- init state ----
  if (tid < NS) s_mu[tid] = mu0[tid];
  for (int idx = tid; idx < NS * NS; idx += NTH) s_Sig[idx] = Sig0[idx];
  __syncthreads();

  // =========================== FORWARD FILTER ===========================
  for (int t = 0; t < TT; ++t) {
    const size_t bt = (size_t)b * TT + t;
    const float* Abt = Ag + bt * (NS * NS);
    const float* Bbt = Bg + bt * (NS * MD);
    const float* Cbt = Cg + bt * (PD * NS);
    for (int idx = tid; idx < NS * NS; idx += NTH) s_A[idx] = Abt[idx];
    for (int idx = tid; idx < NS * MD; idx += NTH) s_Bm[idx] = Bbt[idx];
    for (int idx = tid; idx < PD * NS; idx += NTH) s_C[idx] = Cbt[idx];
    if (tid < PD) s_y[tid] = Yg[bt * PD + tid];
    if (tid < MD) s_u[tid] = Ug[bt * MD + tid];
    __syncthreads();

    // Prefetch next timestep's operands so HBM/L2 latency overlaps this
    // step's barrier-separated compute phases (serial-chain latency hiding).
    if (t + 1 < TT) {
      const size_t btn = bt + 1;
      prefetch_bytes(Ag + btn * (NS * NS), NS * NS * 4, tid);
      prefetch_bytes(Cg + btn * (PD * NS), PD * NS * 4, tid);
      prefetch_bytes(Bg + btn * (NS * MD), NS * MD * 4, tid);
      if (tid == 0) {
        __builtin_prefetch(Yg + btn * PD, 0, 1);
        __builtin_prefetch(Ug + btn * MD, 0, 1);
      }
    }

    // mu_p = A mu + B u ; T1 = A @ Sig
    if (tid < NS) {
      float s = 0.f;
      for (int j = 0; j < NS; ++j) s += s_A[tid * NS + j] * s_mu[j];
      for (int j = 0; j < MD; ++j) s += s_Bm[tid * MD + j] * s_u[j];
      s_mup[tid] = s;
    }
    wmma_mm(s_A, NS, s_Sig, NS, false, s_T1, NS, NS, NS, NS, tid);
    __syncthreads();

    // Sig_p = T1 @ A^T (+Q) ; r = y - C mu_p
    wmma_mm(s_T1, NS, s_A, NS, true, s_Sigp, NS, NS, NS, NS, tid);
    if (tid < PD) {
      float s = s_y[tid];
      for (int j = 0; j < NS; ++j) s -= s_C[tid * NS + j] * s_mup[j];
      s_r[tid] = s;
    }
    __syncthreads();
    if (tid < NS) s_Sigp[tid * NS + tid] += QV;
    __syncthreads();

    // T2 = C @ Sig_p (16x32)
    wmma_mm(s_C, NS, s_Sigp, NS, false, s_T2, NS, PD, NS, NS, tid);
    __syncthreads();
    // S = T2 @ C^T (16x16)
    wmma_mm(s_T2, NS, s_C, NS, true, s_S, PD, PD, PD, NS, tid);
    __syncthreads();
    if (tid < PD) s_S[tid * PD + tid] += RV;
    __syncthreads();
    // S = sym(S)
    {
      int i0 = tid >> 4, j0 = tid & 15;
      float v0 = 0.5f * (s_S[i0 * PD + j0] + s_S[j0 * PD + i0]);
      int idx1 = tid + NTH; int i1 = idx1 >> 4, j1 = idx1 & 15;
      float v1 = 0.5f * (s_S[i1 * PD + j1] + s_S[j1 * PD + i1]);
      __syncthreads();
      s_S[tid] = v0; s_S[tid + NTH] = v1;
    }
    __syncthreads();

    // Cholesky S = L L^T ; solve S X = T2 (32 RHS columns) ; K = X^T
    chol_inplace(s_S, PD, tid);
    if (tid < NS) {
      for (int i = 0; i < PD; ++i) {
        float s = s_T2[i * NS + tid];
        for (int j = 0; j < i; ++j) s -= s_S[i * PD + j] * s_X16[j * NS + tid];
        s_X16[i * NS + tid] = s / s_S[i * PD + i];
      }
      for (int i = PD - 1; i >= 0; --i) {
        float s = s_X16[i * NS + tid];
        for (int j = i + 1; j < PD; ++j) s -= s_S[j * PD + i] * s_X16[j * NS + tid];
        s_X16[i * NS + tid] = s / s_S[i * PD + i];
      }
    }
    __syncthreads();
    for (int idx = tid; idx < NS * PD; idx += NTH) {
      int i2 = idx >> 4, j2 = idx & 15;          // K[i2][j2] = X[j2][i2]
      s_K[idx] = s_X16[j2 * NS + i2];
    }
    __syncthreads();

    // mu_f = mu_p + K r  ;  IKC_raw = K @ C
    if (tid < NS) {
      float s = s_mup[tid];
      for (int j = 0; j < PD; ++j) s += s_K[tid * PD + j] * s_r[j];
      s_mu[tid] = s;
    }
    wmma_mm(s_K, PD, s_C, NS, false, s_IKC, NS, NS, NS, PD, tid);
    __syncthreads();
    // IKC = I - K C  ;  KKT = K @ K^T
    for (int idx = tid; idx < NS * NS; idx += NTH) {
      int i2 = idx >> 5, j2 = idx & 31;
      s_IKC[idx] = (i2 == j2 ? 1.0f : 0.0f) - s_IKC[idx];
    }
    wmma_mm(s_K, PD, s_K, PD, true, s_KKT, NS, NS, NS, PD, tid);
    __syncthreads();
    // T3 = IKC @ Sig_p
    wmma_mm(s_IKC, NS, s_Sigp, NS, false, s_T3, NS, NS, NS, NS, tid);
    __syncthreads();
    // T1 = T3 @ IKC^T
    wmma_mm(s_T3, NS, s_IKC, NS, true, s_T1, NS, NS, NS, NS, tid);
    __syncthreads();

    // Sig_f = sym(T1 + RV * KKT) -> s_Sig ; store filtered state into out slot
    float* outbt = out + bt * (NS * (NS + 1));
    for (int idx = tid; idx < NS * NS; idx += NTH) {
      int i2 = idx >> 5, j2 = idx & 31;
      float a01 = s_T1[idx] + RV * s_KKT[idx];
      float a10 = s_T1[j2 * NS + i2] + RV * s_KKT[j2 * NS + i2];
      float v = 0.5f * (a01 + a10);
      s_Sig[idx] = v;
      outbt[i2 * (NS + 1) + 1 + j2] = v;
    }
    if (tid < NS) outbt[tid * (NS + 1)] = s_mu[tid];
    __syncthreads();
  }

  // ======================== BACKWARD RTS SMOOTHER ========================
  // s_mus / s_KKT carry the smoothed state (init = filtered at T-1).
  if (tid < NS) s_mus[tid] = s_mu[tid];
  for (int idx = tid; idx < NS * NS; idx += NTH) s_KKT[idx] = s_Sig[idx];
  __syncthreads();

  for (int t = TT - 2; t >= 0; --t) {
    const size_t bt  = (size_t)b * TT + t;
    const size_t bt1 = bt + 1;
    const float* A1 = Ag + bt1 * (NS * NS);
    const float* B1 = Bg + bt1 * (NS * MD);
    float* outbt = out + bt * (NS * (NS + 1));
    for (int idx = tid; idx < NS * NS; idx += NTH) s_A[idx] = A1[idx];
    for (int idx = tid; idx < NS * MD; idx += NTH) s_Bm[idx] = B1[idx];
    if (tid < MD) s_u[tid] = Ug[bt1 * MD + tid];
    // load filtered state at t from out slot t
    for (int idx = tid; idx < NS * NS; idx += NTH) {
      int i2 = idx >> 5, j2 = idx & 31;
      s_Sig[idx] = outbt[i2 * (NS + 1) + 1 + j2];
    }
    if (tid < NS) s_mu[tid] = outbt[tid * (NS + 1)];
    __syncthreads();

    // Prefetch next backward iteration's operands (A/B at t, out slot t-1).
    if (t > 0) {
      prefetch_bytes(Ag + bt * (NS * NS), NS * NS * 4, tid);
      prefetch_bytes(Bg + bt * (NS * MD), NS * MD * 4, tid);
      prefetch_bytes(out + (bt - 1) * (NS * (NS + 1)), NS * (NS + 1) * 4, tid);
    }

    // mp1 = A1 mf + B1 u1 ; T1 = A1 @ Sf
    if (tid < NS) {
      float s = 0.f;
      for (int j = 0; j < NS; ++j) s += s_A[tid * NS + j] * s_mu[j];
      for (int j = 0; j < MD; ++j) s += s_Bm[tid * MD + j] * s_u[j];
      s_mp1[tid] = s;
    }
    wmma_mm(s_A, NS, s_Sig, NS, false, s_T1, NS, NS, NS, NS, tid);
    __syncthreads();
    // Sp1 = T1 @ A1^T (+Q) ; G = Sf @ A1^T
    wmma_mm(s_T1, NS, s_A, NS, true, s_Sigp, NS, NS, NS, NS, tid);
    wmma_mm(s_Sig, NS, s_A, NS, true, s_T3, NS, NS, NS, NS, tid);
    __syncthreads();
    if (tid < NS) s_Sigp[tid * NS + tid] += QV;
    __syncthreads();

    // Cholesky of a copy of Sp1 ; solve Sp1 W = G^T  (column tid)
    for (int idx = tid; idx < NS * NS; idx += NTH) s_T1[idx] = s_Sigp[idx];
    __syncthreads();
    chol_inplace(s_T1, NS, tid);
    if (tid < NS) {
      for (int i = 0; i < NS; ++i) {
        float s = s_T3[tid * NS + i];              // G^T[i][tid] = G[tid][i]
        for (int j = 0; j < i; ++j) s -= s_T1[i * NS + j] * s_IKC[j * NS + tid];
        s_IKC[i * NS + tid] = s / s_T1[i * NS + i];
      }
      for (int i = NS - 1; i >= 0; --i) {
        float s = s_IKC[i * NS + tid];
        for (int j = i + 1; j < NS; ++j) s -= s_T1[j * NS + i] * s_IKC[j * NS + tid];
        s_IKC[i * NS + tid] = s / s_T1[i * NS + i];
      }
    }
    __syncthreads();

    // J = W^T -> s_T3 (G dead) ; D = Sig_s - Sp1 -> s_KKT in place
    for (int idx = tid; idx < NS * NS; idx += NTH) {
      int i2 = idx >> 5, j2 = idx & 31;
      s_T3[idx] = s_IKC[j2 * NS + i2];
    }
    for (int idx = tid; idx < NS * NS; idx += NTH) s_KKT[idx] -= s_Sigp[idx];
    __syncthreads();

    // mu_n = mf + J (mu_s - mp1) ; JD = J @ D
    if (tid < NS) {
      float s = s_mu[tid];
      for (int j = 0; j < NS; ++j) s += s_T3[tid * NS + j] * (s_mus[j] - s_mp1[j]);
      s_mup[tid] = s;
    }
    wmma_mm(s_T3, NS, s_KKT, NS, false, s_JD, NS, NS, NS, NS, tid);
    __syncthreads();
    // T1 = JD @ J^T
    wmma_mm(s_JD, NS, s_T3, NS, true, s_T1, NS, NS, NS, NS, tid);
    __syncthreads();

    // Sig_n = Sf + sym(T1) -> s_KKT (new smoothed) + out ; mu update
    for (int idx = tid; idx < NS * NS; idx += NTH) {
      int i2 = idx >> 5, j2 = idx & 31;
      float v = s_Sig[idx] + 0.5f * (s_T1[idx] + s_T1[j2 * NS + i2]);
      s_KKT[idx] = v;
      outbt[i2 * (NS + 1) + 1 + j2] = v;
    }
    if (tid < NS) {
      s_mus[tid] = s_mup[tid];
      outbt[tid * (NS + 1)] = s_mup[tid];
    }
    __syncthreads();
  }
}

extern "C" void kernel_launch(void* const* d_in, const int* in_sizes, int n_in,
                              void* d_out, int out_size, void* d_ws, size_t ws_size,
                              hipStream_t stream) {
  (void)in_sizes; (void)n_in; (void)out_size; (void)d_ws; (void)ws_size;
  const float* Yg   = (const float*)d_in[0];
  const float* Ug   = (const float*)d_in[1];
  const float* Ag   = (const float*)d_in[2];
  const float* Bg   = (const float*)d_in[3];
  const float* Cg   = (const float*)d_in[4];
  const float* mu0  = (const float*)d_in[5];
  const float* Sig0 = (const float*)d_in[6];
  float* outp = (float*)d_out;
  hipLaunchKernelGGL(kalman_smooth_kernel, dim3(NB), dim3(NTH), 0, stream,
                     Yg, Ug, Ag, Bg, Cg, mu0, Sig0, outp);
}